// SampleTopKPatch_31920196944415
// MI455X (gfx1250) — compile-verified
//
#include <hip/hip_runtime.h>
#include <stdint.h>

typedef __attribute__((ext_vector_type(2))) float v2f;
typedef __attribute__((ext_vector_type(4))) float v4f;
typedef __attribute__((ext_vector_type(8))) float v8f;

#define K_SEL      8
#define N_PATCH    64
#define TAU_INV    100.0f
#define NEGV       (-1e9f)
// image geometry: [1,4,256,256,256], patches 64^3, grid 4x4x4
#define IMG_C_STRIDE 16777216   // 256^3
#define IMG_H_STRIDE 65536      // 256^2
#define IMG_W_STRIDE 256
#define IMG_M        1048576    // 4*64^3 per-k image out elements
#define LAB_M        262144     // 64^3 per-k label out elements

// ---------------------------------------------------------------------------
// Kernel 1: masked logit + iterative Gumbel top-k (tiny; one thread).
// Writes masked_logit (64 f32) to out_logit, and (idx[k], s[k]) to workspace.
// ---------------------------------------------------------------------------
__global__ void topk_kernel(const float* __restrict__ logit,
                            const float* __restrict__ bg,
                            const float* __restrict__ gum,
                            float* __restrict__ out_logit,
                            float* __restrict__ ws_s,
                            int*   __restrict__ ws_idx) {
  float pert[N_PATCH], mask[N_PATCH];
  for (int i = 0; i < N_PATCH; ++i) {
    float lm = logit[i] + logf(fmaxf(1.0f - bg[i], 1e-8f));
    out_logit[i] = lm;
    pert[i] = lm + gum[i];
    mask[i] = 0.0f;
  }
  for (int k = 0; k < K_SEL; ++k) {
    float best = -3.4e38f; int bi = 0;
    for (int i = 0; i < N_PATCH; ++i) {
      float x = pert[i] + mask[i];
      if (x > best) { best = x; bi = i; }   // first-max, matches jnp.argmax
    }
    float sum = 0.0f;
    for (int i = 0; i < N_PATCH; ++i)
      sum += expf((pert[i] + mask[i] - best) * TAU_INV);
    ws_s[k] = 1.0f / sum;                   // softmax value at argmax
    ws_idx[k] = bi;
    mask[bi] += NEGV;
  }
}

// ---------------------------------------------------------------------------
// Kernel 2: image gather+scale as the reference einsum on tensor cores.
// out[16xM] = diag16(s) [16x8] x Psel [8xM] via V_WMMA_F32_16X16X4_F32,
// two K-chunks of 4; rows 8..15 are zero and not stored.
// Each wave iteration handles one (c,hp,wp) row = 64 contiguous dp values.
// ---------------------------------------------------------------------------
__device__ __forceinline__ size_t patch_origin(int n) {
  int hn = n >> 4, wn = (n >> 2) & 3, dn = n & 3;
  return (size_t)(hn * 64) * IMG_H_STRIDE + (size_t)(wn * 64) * IMG_W_STRIDE +
         (size_t)(dn * 64);
}

__global__ __launch_bounds__(256) void img_wmma_kernel(
    const float* __restrict__ img, float* __restrict__ out,
    const float* __restrict__ ws_s, const int* __restrict__ ws_idx) {
  const int lane = threadIdx.x & 31;
  const int half = lane >> 4;     // lanes 0-15 vs 16-31
  const int L    = lane & 15;
  const int j0   = half * 2;      // K-slot pair owned by this half (ISA A/B layout)

  // A = diag(s) padded to 16 rows; per-lane two K-slots per chunk.
  float wl = (L < K_SEL) ? ws_s[L] : 0.0f;
  v2f a0, a1;
  a0.x = (L == j0    ) ? wl : 0.0f;   // chunk0: K = j0
  a0.y = (L == j0 + 1) ? wl : 0.0f;   // chunk0: K = j0+1
  a1.x = (L == j0 + 4) ? wl : 0.0f;   // chunk1: K = 4+j0
  a1.y = (L == j0 + 5) ? wl : 0.0f;   // chunk1: K = 5+j0

  // B rows this lane supplies: selected patches j0, j0+1 (chunk0), +4 (chunk1)
  const size_t oA = patch_origin(ws_idx[j0]);
  const size_t oB = patch_origin(ws_idx[j0 + 1]);
  const size_t oC = patch_origin(ws_idx[j0 + 4]);
  const size_t oD = patch_origin(ws_idx[j0 + 5]);

  const int wid    = blockIdx.x * (blockDim.x >> 5) + (threadIdx.x >> 5);
  const int nwaves = gridDim.x * (blockDim.x >> 5);

  for (int blk = wid; blk < 4 * 64 * 64; blk += nwaves) {
    const int ch = blk >> 12;
    const int hp = (blk >> 6) & 63;
    const int wp = blk & 63;
    const size_t colOff = (size_t)ch * IMG_C_STRIDE + (size_t)hp * IMG_H_STRIDE +
                          (size_t)wp * IMG_W_STRIDE + 4 * L;

    const v4f fa = *(const v4f*)(img + oA + colOff);
    const v4f fb = *(const v4f*)(img + oB + colOff);
    const v4f fc = *(const v4f*)(img + oC + colOff);
    const v4f fd = *(const v4f*)(img + oD + colOff);

    const v8f z = {0.f,0.f,0.f,0.f,0.f,0.f,0.f,0.f};
    v8f d0, d1, d2, d3;
    v2f b;
    // sub-tile t covers dp = 4*L + t
    b.x = fa.x; b.y = fb.x;
    d0 = __builtin_amdgcn_wmma_f32_16x16x4_f32(false, a0, false, b, (short)0, z,  false, false);
    b.x = fc.x; b.y = fd.x;
    d0 = __builtin_amdgcn_wmma_f32_16x16x4_f32(false, a1, false, b, (short)0, d0, false, false);

    b.x = fa.y; b.y = fb.y;
    d1 = __builtin_amdgcn_wmma_f32_16x16x4_f32(false, a0, false, b, (short)0, z,  false, false);
    b.x = fc.y; b.y = fd.y;
    d1 = __builtin_amdgcn_wmma_f32_16x16x4_f32(false, a1, false, b, (short)0, d1, false, false);

    b.x = fa.z; b.y = fb.z;
    d2 = __builtin_amdgcn_wmma_f32_16x16x4_f32(false, a0, false, b, (short)0, z,  false, false);
    b.x = fc.z; b.y = fd.z;
    d2 = __builtin_amdgcn_wmma_f32_16x16x4_f32(false, a1, false, b, (short)0, d2, false, false);

    b.x = fa.w; b.y = fb.w;
    d3 = __builtin_amdgcn_wmma_f32_16x16x4_f32(false, a0, false, b, (short)0, z,  false, false);
    b.x = fc.w; b.y = fd.w;
    d3 = __builtin_amdgcn_wmma_f32_16x16x4_f32(false, a1, false, b, (short)0, d3, false, false);

    // D rows 0..7 live in lanes 0-15 of VGPRs 0..7; pack 4 dp values -> b128 store
    if (half == 0) {
      const size_t mO = (size_t)ch * LAB_M + (size_t)hp * 4096 + (size_t)wp * 64 + 4 * L;
#pragma unroll
      for (int v = 0; v < K_SEL; ++v) {
        v4f o;
        o.x = d0[v]; o.y = d1[v]; o.z = d2[v]; o.w = d3[v];
        *(v4f*)(out + (size_t)v * IMG_M + mO) = o;
      }
    }
  }
}

// ---------------------------------------------------------------------------
// Kernel 3: label gather as a pure global->LDS->global async DMA copy.
// 524288 float4s total; 32768 threads x 4 in-flight x 4 passes.
// ---------------------------------------------------------------------------
#define LAB_THREADS 32768
__global__ __launch_bounds__(256) void lab_async_kernel(
    const float* __restrict__ lab, float* __restrict__ out,
    const int* __restrict__ ws_idx) {
  __shared__ __align__(16) char stage[256 * 4 * 16];

  const int tid = blockIdx.x * blockDim.x + threadIdx.x;
  // low 32 bits of a flat shared-memory address == LDS byte offset
  const unsigned ldsBase =
      (unsigned)(unsigned long long)(uintptr_t)(void*)&stage[threadIdx.x * 64];

  int idx8[K_SEL];
#pragma unroll
  for (int i = 0; i < K_SEL; ++i) idx8[i] = ws_idx[i];

  for (int pass = 0; pass < 4; ++pass) {
    unsigned long long dsts[4];
#pragma unroll
    for (int d = 0; d < 4; ++d) {
      const int g = tid + (pass * 4 + d) * LAB_THREADS;   // float4 index
      const int k = g >> 16;                 // 65536 float4 per k
      const int m = (g & 65535) << 2;        // element offset within patch
      const int n = idx8[k];
      const int hn = n >> 4, wn = (n >> 2) & 3, dn = n & 3;
      const int hp = m >> 12, wp = (m >> 6) & 63, dp = m & 63;
      const size_t inElem = (size_t)(hn * 64 + hp) * IMG_H_STRIDE +
                            (size_t)(wn * 64 + wp) * IMG_W_STRIDE +
                            (size_t)(dn * 64 + dp);
      const unsigned long long src = (unsigned long long)(uintptr_t)(lab + inElem);
      dsts[d] = (unsigned long long)(uintptr_t)(out + (size_t)k * LAB_M + m);
      const unsigned ldsOff = ldsBase + d * 16;
      asm volatile("global_load_async_to_lds_b128 %0, %1, off"
                   :: "v"(ldsOff), "v"(src) : "memory");
    }
    asm volatile("s_wait_asynccnt 0" ::: "memory");
#pragma unroll
    for (int d = 0; d < 4; ++d) {
      const unsigned ldsOff = ldsBase + d * 16;
      asm volatile("global_store_async_from_lds_b128 %0, %1, off"
                   :: "v"(dsts[d]), "v"(ldsOff) : "memory");
    }
    asm volatile("s_wait_asynccnt 0" ::: "memory");
  }
}

// ---------------------------------------------------------------------------
extern "C" void kernel_launch(void* const* d_in, const int* in_sizes, int n_in,
                              void* d_out, int out_size, void* d_ws, size_t ws_size,
                              hipStream_t stream) {
  const float* img   = (const float*)d_in[0];  // [1,4,256,256,256]
  const float* lab   = (const float*)d_in[1];  // [1,1,256,256,256]
  const float* logit = (const float*)d_in[2];  // 64
  const float* bg    = (const float*)d_in[3];  // 64
  const float* gum   = (const float*)d_in[4];  // 64
  (void)in_sizes; (void)n_in; (void)out_size; (void)ws_size;

  float* out       = (float*)d_out;
  float* out_img   = out;                                   // 8*4*64^3
  float* out_lab   = out + (size_t)K_SEL * IMG_M;           // 8*64^3
  float* out_logit = out_lab + (size_t)K_SEL * LAB_M;       // 64

  float* ws_s  = (float*)d_ws;
  int*   ws_idx = (int*)(ws_s + K_SEL);

  topk_kernel<<<1, 1, 0, stream>>>(logit, bg, gum, out_logit, ws_s, ws_idx);
  img_wmma_kernel<<<256, 256, 0, stream>>>(img, out_img, ws_s, ws_idx);
  lab_async_kernel<<<128, 256, 0, stream>>>(lab, out_lab, ws_idx);
}